// PatchDensityVariance_90855738179773
// MI455X (gfx1250) — compile-verified
//
#include <hip/hip_runtime.h>

#define N_PTS 65536
#define N_B   16
#define N_S   40
#define R2    0.0025f

typedef __attribute__((ext_vector_type(2))) float v2f;
typedef __attribute__((ext_vector_type(8))) float v8f;

// ---------------------------------------------------------------------------
// Phase 1: farthest point sampling. One workgroup per batch (16 WGs x 1024
// threads). Running `distance` lives in global scratch (L2-resident, 4MB).
// Seed i recorded is the centroid *used* at step i (matches the lax.scan
// emitting the pre-update `farthest`, with far0 = 0).
// ---------------------------------------------------------------------------
__global__ __launch_bounds__(1024) void fps_kernel(const float* __restrict__ pcs,
                                                   float* __restrict__ dist,
                                                   float* __restrict__ seed_xyz) {
  const int b = blockIdx.x;
  const float* pb = pcs + (size_t)b * N_PTS * 3;
  float* db = dist + (size_t)b * N_PTS;

  __shared__ float s_val[32];
  __shared__ int   s_idx[32];
  __shared__ float s_c[3];
  __shared__ int   s_far;

  const int tid  = threadIdx.x;
  const int lane = tid & 31;
  const int wave = tid >> 5;

  int far = 0;
  for (int it = 0; it < N_S; ++it) {
    if (tid == 0) {
      float cx = pb[3 * far + 0];
      float cy = pb[3 * far + 1];
      float cz = pb[3 * far + 2];
      s_c[0] = cx; s_c[1] = cy; s_c[2] = cz;
      float* s = seed_xyz + (size_t)(b * N_S + it) * 3;
      s[0] = cx; s[1] = cy; s[2] = cz;
    }
    __syncthreads();
    const float cx = s_c[0], cy = s_c[1], cz = s_c[2];

    float bestv = -1.0f;
    int   besti = 0;
    {
#pragma clang fp contract(off)
      for (int i = tid; i < N_PTS; i += 1024) {
        float dx = pb[3 * i + 0] - cx;
        float dy = pb[3 * i + 1] - cy;
        float dz = pb[3 * i + 2] - cz;
        float d  = dx * dx + dy * dy + dz * dz;
        float old = (it == 0) ? 1e10f : db[i];
        float nd  = fminf(old, d);
        db[i] = nd;
        if (nd > bestv || (nd == bestv && i < besti)) { bestv = nd; besti = i; }
      }
    }
    // wave32 butterfly reduce: (max value, min index on ties) == jnp.argmax
    for (int m = 16; m >= 1; m >>= 1) {
      float ov = __shfl_xor(bestv, m, 32);
      int   oi = __shfl_xor(besti, m, 32);
      if (ov > bestv || (ov == bestv && oi < besti)) { bestv = ov; besti = oi; }
    }
    if (lane == 0) { s_val[wave] = bestv; s_idx[wave] = besti; }
    __syncthreads();
    if (wave == 0) {
      float v  = s_val[lane];
      int   ix = s_idx[lane];
      for (int m = 16; m >= 1; m >>= 1) {
        float ov = __shfl_xor(v, m, 32);
        int   oi = __shfl_xor(ix, m, 32);
        if (ov > v || (ov == v && oi < ix)) { v = ov; ix = oi; }
      }
      if (lane == 0) s_far = ix;
    }
    __syncthreads();
    far = s_far;
  }
}

// ---------------------------------------------------------------------------
// Zero den_mat (harness poisons d_ws once; we must re-init every call).
// ---------------------------------------------------------------------------
__global__ void zero_kernel(float* __restrict__ den_mat, int n) {
  int i = blockIdx.x * blockDim.x + threadIdx.x;
  if (i < n) den_mat[i] = 0.0f;
}

// ---------------------------------------------------------------------------
// Phase 2: den_mat[b][s] = sum_n relu(R^2 - (|s|^2 + |p|^2 - 2 s.p)) using
// V_WMMA_F32_16X16X4_F32 with augmented operands:
//   A row m   = (-2sx, -2sy, -2sz, 1)        (16x4, f32)
//   B col n   = ( px,   py,   pz, |p|^2)     (4x16, f32)
//   D[m][n]   = |p|^2 - 2 s.p
//   relu term = max(0, (R^2 - |s|^2) - D[m][n])
// Grid: x = 48 (batch * seed-tile), y = 8 (point chunk). Block = 256 (8 waves).
// ---------------------------------------------------------------------------
__global__ __launch_bounds__(256) void density_kernel(const float* __restrict__ pcs,
                                                      const float* __restrict__ seed_xyz,
                                                      float* __restrict__ den_mat) {
  const int bt    = blockIdx.x;   // 0..47
  const int b     = bt / 3;
  const int st    = bt % 3;       // seed tile (16 rows)
  const int chunk = blockIdx.y;   // 0..7

  __shared__ float aug[16][4];    // augmented A rows
  __shared__ float rc[16];        // R^2 - |s|^2 per row (pad rows: -1 -> relu 0)

  const int tid = threadIdx.x;
  if (tid < 16) {
    const int sg = st * 16 + tid;
    float sx = 0.f, sy = 0.f, sz = 0.f, one = 0.f, r = -1.0f;
    if (sg < N_S) {
      const float* s = seed_xyz + (size_t)(b * N_S + sg) * 3;
      sx = s[0]; sy = s[1]; sz = s[2];
      one = 1.0f;
      r = R2 - (sx * sx + sy * sy + sz * sz);
    }
    aug[tid][0] = -2.0f * sx;
    aug[tid][1] = -2.0f * sy;
    aug[tid][2] = -2.0f * sz;
    aug[tid][3] = one;
    rc[tid] = r;
  }
  __syncthreads();

  const int lane = tid & 31;
  const int wave = tid >> 5;
  const int half = lane >> 4;     // 0: K=0,1 / rows 0..7 ; 1: K=2,3 / rows 8..15
  const int l15  = lane & 15;

  // A operand: 16x4 f32 layout — lanes 0-15 hold K=0(V0),K=1(V1);
  // lanes 16-31 hold K=2(V0),K=3(V1), same rows M=0..15.
  v2f A;
  A[0] = aug[l15][half ? 2 : 0];
  A[1] = aug[l15][half ? 3 : 1];

  float rcv[8];
#pragma unroll
  for (int v = 0; v < 8; ++v) rcv[v] = rc[v + 8 * half];

  v8f acc = {};
  const float* pb = pcs + (size_t)b * N_PTS * 3;
  const int tilesPerChunk = (N_PTS / 16) / 8;   // 512 tiles of 16 points
  const int t0 = chunk * tilesPerChunk;

  for (int t = t0 + wave; t < t0 + tilesPerChunk; t += 8) {
    const int n = t * 16 + l15;
    const float px = pb[3 * n + 0];
    const float py = pb[3 * n + 1];
    const float pz = pb[3 * n + 2];
    // B operand: 4x16 f32 — lanes 0-15: K=0(V0)=px, K=1(V1)=py;
    // lanes 16-31: K=2(V0)=pz, K=3(V1)=|p|^2.
    v2f B;
    if (half == 0) { B[0] = px; B[1] = py; }
    else           { B[0] = pz; B[1] = px * px + py * py + pz * pz; }

    v8f c = {};
    v8f D = __builtin_amdgcn_wmma_f32_16x16x4_f32(
        /*neg_a=*/false, A, /*neg_b=*/false, B,
        /*c_mod=*/(short)0, c, /*reuse_a=*/false, /*reuse_b=*/false);

#pragma unroll
    for (int v = 0; v < 8; ++v) acc[v] += fmaxf(0.0f, rcv[v] - D[v]);
  }

  // Sum each row across the 16 lanes of its half (xor <= 8 stays in half).
#pragma unroll
  for (int m = 1; m <= 8; m <<= 1) {
#pragma unroll
    for (int v = 0; v < 8; ++v) acc[v] += __shfl_xor(acc[v], m, 32);
  }

  if (l15 == 0) {
#pragma unroll
    for (int v = 0; v < 8; ++v) {
      const int mrow = v + 8 * half;        // D layout: VGPR v -> rows v / v+8
      const int sg = st * 16 + mrow;
      if (sg < N_S) atomicAdd(&den_mat[b * N_S + sg], acc[v]);
    }
  }
}

// ---------------------------------------------------------------------------
// Phase 3: den_var = mean_b( var(den_mat[b,:], ddof=1) )
// ---------------------------------------------------------------------------
__global__ void var_kernel(const float* __restrict__ den_mat, float* __restrict__ out) {
  __shared__ float pv[16];
  const int b = threadIdx.x;
  if (b < N_B) {
    float s = 0.0f;
    for (int j = 0; j < N_S; ++j) s += den_mat[b * N_S + j];
    const float mean = s / (float)N_S;
    float ss = 0.0f;
    for (int j = 0; j < N_S; ++j) {
      const float d = den_mat[b * N_S + j] - mean;
      ss += d * d;
    }
    pv[b] = ss / (float)(N_S - 1);
  }
  __syncthreads();
  if (threadIdx.x == 0) {
    float t = 0.0f;
    for (int i = 0; i < N_B; ++i) t += pv[i];
    out[0] = t / (float)N_B;
  }
}

extern "C" void kernel_launch(void* const* d_in, const int* in_sizes, int n_in,
                              void* d_out, int out_size, void* d_ws, size_t ws_size,
                              hipStream_t stream) {
  (void)in_sizes; (void)n_in; (void)out_size; (void)ws_size;
  const float* pcs = (const float*)d_in[0];
  float* out = (float*)d_out;

  char* ws = (char*)d_ws;
  float* dist     = (float*)ws;                                  // 16*65536 f32 = 4 MB
  float* seed_xyz = (float*)(ws + (size_t)N_B * N_PTS * 4);      // 16*40*3 f32
  float* den_mat  = seed_xyz + N_B * N_S * 3;                    // 16*40 f32

  zero_kernel<<<(N_B * N_S + 255) / 256, 256, 0, stream>>>(den_mat, N_B * N_S);
  fps_kernel<<<N_B, 1024, 0, stream>>>(pcs, dist, seed_xyz);
  density_kernel<<<dim3(48, 8), 256, 0, stream>>>(pcs, seed_xyz, den_mat);
  var_kernel<<<1, 32, 0, stream>>>(den_mat, out);
}